// InteractionGNN_3934190043555
// MI455X (gfx1250) — compile-verified
//
#include <hip/hip_runtime.h>

#define N_NODES 50000
#define N_EDGES 800000
#define D 128

typedef __attribute__((ext_vector_type(2))) float v2f;
typedef __attribute__((ext_vector_type(8))) float v8f;

// ---------------------------------------------------------------------------
// Zero-fill 32-bit words (grid-stride)
// ---------------------------------------------------------------------------
__global__ void zero_b32(unsigned* __restrict__ p, long long n) {
    long long i = (long long)blockIdx.x * blockDim.x + threadIdx.x;
    long long stride = (long long)gridDim.x * blockDim.x;
    for (; i < n; i += stride) p[i] = 0u;
}

// ---------------------------------------------------------------------------
// Degree count (u32 atomics, 800k total -- cheap)
// ---------------------------------------------------------------------------
__global__ void degree_kernel(const long long* __restrict__ dst,
                              unsigned* __restrict__ cnt, int nedges) {
    int e = blockIdx.x * blockDim.x + threadIdx.x;
    if (e < nedges) atomicAdd(&cnt[(int)dst[e]], 1u);
}

// ---------------------------------------------------------------------------
// Two-level exclusive scan of cnt[] -> rowptr[] (CSR offsets)
// scan1: per-block exclusive scan (LDS Hillis-Steele), block sums -> bsum
// ---------------------------------------------------------------------------
__global__ __launch_bounds__(256)
void scan1_kernel(const unsigned* __restrict__ cnt,
                  unsigned* __restrict__ excl,
                  unsigned* __restrict__ bsum, int n) {
    __shared__ unsigned s[256];
    int tid = threadIdx.x;
    int i = blockIdx.x * 256 + tid;
    unsigned v0 = (i < n) ? cnt[i] : 0u;
    s[tid] = v0;
    __syncthreads();
#pragma unroll
    for (int off = 1; off < 256; off <<= 1) {
        unsigned add = (tid >= off) ? s[tid - off] : 0u;
        __syncthreads();
        s[tid] += add;
        __syncthreads();
    }
    if (i < n) excl[i] = s[tid] - v0;
    if (tid == 255) bsum[blockIdx.x] = s[255];
}

// scan2: single block, exclusive scan of nb (<256) block sums in place
__global__ __launch_bounds__(256)
void scan2_kernel(unsigned* __restrict__ bsum, int nb) {
    __shared__ unsigned s[256];
    int tid = threadIdx.x;
    unsigned v0 = (tid < nb) ? bsum[tid] : 0u;
    s[tid] = v0;
    __syncthreads();
#pragma unroll
    for (int off = 1; off < 256; off <<= 1) {
        unsigned add = (tid >= off) ? s[tid - off] : 0u;
        __syncthreads();
        s[tid] += add;
        __syncthreads();
    }
    if (tid < nb) bsum[tid] = s[tid] - v0;
}

// scan3: add block offsets; duplicate into cursor for the CSR fill
__global__ void scan3_kernel(unsigned* __restrict__ rowptr,
                             const unsigned* __restrict__ bsum,
                             unsigned* __restrict__ cursor, int n) {
    int i = blockIdx.x * blockDim.x + threadIdx.x;
    if (i < n) {
        unsigned r = rowptr[i] + bsum[blockIdx.x];
        rowptr[i] = r;
        cursor[i] = r;
    }
}

// ---------------------------------------------------------------------------
// CSR fill (counting sort of edges by dst): colidx[slot] = src
// ---------------------------------------------------------------------------
__global__ void csr_fill_kernel(const long long* __restrict__ src,
                                const long long* __restrict__ dst,
                                unsigned* __restrict__ cursor,
                                int* __restrict__ colidx, int nedges) {
    int e = blockIdx.x * blockDim.x + threadIdx.x;
    if (e < nedges) {
        unsigned pos = atomicAdd(&cursor[(int)dst[e]], 1u);
        colidx[pos] = (int)src[e];
    }
}

// ---------------------------------------------------------------------------
// Atomic-free gather aggregation: one wave per node. The 32 lanes read a full
// 512B neighbor feature row per iteration (coalesced float4), accumulate in
// registers, and write the MEAN directly (no separate scale pass).
// ---------------------------------------------------------------------------
__global__ __launch_bounds__(256)
void aggregate_mean_kernel(const float* __restrict__ feat,
                           const unsigned* __restrict__ rowptr,
                           const unsigned* __restrict__ cnt,
                           const int* __restrict__ colidx,
                           float* __restrict__ mean, int nnodes) {
    const int lane = threadIdx.x & 31;
    const int wave = threadIdx.x >> 5;
    const int i = blockIdx.x * 8 + wave;
    if (i >= nnodes) return;
    const unsigned start = rowptr[i];
    const unsigned deg = cnt[i];

    float a0 = 0.f, a1 = 0.f, a2 = 0.f, a3 = 0.f;
    for (unsigned p = 0; p < deg; ++p) {
        int j = colidx[start + p];   // wave-uniform -> scalar/broadcast load
        float4 v = *(const float4*)(feat + (size_t)j * D + lane * 4);
        a0 += v.x; a1 += v.y; a2 += v.z; a3 += v.w;
    }
    const float s = 1.0f / fmaxf((float)deg, 1.0f);
    float4 o; o.x = a0 * s; o.y = a1 * s; o.z = a2 * s; o.w = a3 * s;
    *(float4*)(mean + (size_t)i * D + lane * 4) = o;
}

// ---------------------------------------------------------------------------
// Fused SAGE layer GEMM: out = mean @ Wl + bias + x @ Wr (+ optional ReLU).
// One wave per 16-row strip, 8 C-tiles spanning 128 cols, fp32 WMMA 16x16x4.
// ---------------------------------------------------------------------------
__global__ __launch_bounds__(256)
void sage_gemm_kernel(const float* __restrict__ mean,
                      const float* __restrict__ xr,
                      const float* __restrict__ Wl,
                      const float* __restrict__ bias,
                      const float* __restrict__ Wr,
                      float* __restrict__ out,
                      int nrows, int do_relu) {
    const int lane = threadIdx.x & 31;
    const int wave = threadIdx.x >> 5;
    const int rowTile = blockIdx.x * 8 + wave;   // uniform across wave
    const int row0 = rowTile * 16;
    if (row0 >= nrows) return;                   // whole-wave exit, EXEC stays full

    const int lmod = lane & 15;                  // col within tile / A-row
    const int lhi  = lane >> 4;                  // K-half selector

    int arow = row0 + lmod;
    int arowc = arow < nrows ? arow : nrows - 1; // clamp reads in ragged tail
    const float* mRow = mean + (size_t)arowc * D;
    const float* xRow = xr   + (size_t)arowc * D;

    v8f c[8];
#pragma unroll
    for (int t = 0; t < 8; ++t)
#pragma unroll
        for (int e = 0; e < 8; ++e) c[t][e] = 0.0f;

    for (int kb = 0; kb < D; kb += 4) {
        const int ka = kb + 2 * lhi;             // this lane's two consecutive k
        v2f aM, aX;
        {
            float2 m2 = *(const float2*)(mRow + ka);
            float2 x2 = *(const float2*)(xRow + ka);
            aM.x = m2.x; aM.y = m2.y;
            aX.x = x2.x; aX.y = x2.y;
        }
#pragma unroll
        for (int t = 0; t < 8; ++t) {
            const int n = t * 16 + lmod;
            v2f bL, bR;
            bL.x = Wl[(size_t)ka * D + n];
            bL.y = Wl[(size_t)(ka + 1) * D + n];
            bR.x = Wr[(size_t)ka * D + n];
            bR.y = Wr[(size_t)(ka + 1) * D + n];
            c[t] = __builtin_amdgcn_wmma_f32_16x16x4_f32(
                false, aM, false, bL, (short)0, c[t], false, false);
            c[t] = __builtin_amdgcn_wmma_f32_16x16x4_f32(
                false, aX, false, bR, (short)0, c[t], false, false);
        }
    }

    // Epilogue. C layout: elem e, lane l -> row row0+e+8*(l>>4), col t*16+(l&15).
    const bool fullTile = (row0 + 16 <= nrows);  // wave-uniform branch
    if (fullTile) {
#pragma unroll
        for (int t = 0; t < 8; ++t) {
            const int n = t * 16 + lmod;
            const float bv = bias[n];
#pragma unroll
            for (int e = 0; e < 8; ++e) {
                const int r = row0 + e + 8 * lhi;
                float v = c[t][e] + bv;
                if (do_relu) v = fmaxf(v, 0.0f);
                out[(size_t)r * D + n] = v;
            }
        }
    } else {
#pragma unroll
        for (int t = 0; t < 8; ++t) {
            const int n = t * 16 + lmod;
            const float bv = bias[n];
#pragma unroll
            for (int e = 0; e < 8; ++e) {
                const int r = row0 + e + 8 * lhi;
                if (r < nrows) {
                    float v = c[t][e] + bv;
                    if (do_relu) v = fmaxf(v, 0.0f);
                    out[(size_t)r * D + n] = v;
                }
            }
        }
    }
}

// ---------------------------------------------------------------------------
// Host-side orchestration
// ---------------------------------------------------------------------------
extern "C" void kernel_launch(void* const* d_in, const int* in_sizes, int n_in,
                              void* d_out, int out_size, void* d_ws, size_t ws_size,
                              hipStream_t stream) {
    const float*     x    = (const float*)d_in[0];
    const long long* ei   = (const long long*)d_in[1];   // int64 [2, N_EDGES]
    const float*     W1l  = (const float*)d_in[2];
    const float*     b1   = (const float*)d_in[3];
    const float*     W1r  = (const float*)d_in[4];
    const float*     W2l  = (const float*)d_in[5];
    const float*     b2   = (const float*)d_in[6];
    const float*     W2r  = (const float*)d_in[7];
    const long long* src  = ei;
    const long long* dstp = ei + N_EDGES;
    float* out = (float*)d_out;

    // Workspace layout (32-bit words), ~55 MB total:
    //   cnt | rowptr | cursor | bsum | colidx | mean | h
    unsigned* ws      = (unsigned*)d_ws;
    const int NPAD    = 50048;                    // node arrays, 64B-aligned pad
    unsigned* cnt     = ws;
    unsigned* rowptr  = cnt + NPAD;
    unsigned* cursor  = rowptr + NPAD;
    unsigned* bsum    = cursor + NPAD;            // 256 entries
    int*      colidx  = (int*)(bsum + 256);       // N_EDGES
    float*    mean    = (float*)(colidx + N_EDGES);
    float*    h       = mean + (size_t)N_NODES * D;

    const int nb = (N_NODES + 255) / 256;         // 196 scan blocks

    // ---- Build CSR once (shared by both layers) ----
    zero_b32<<<256, 256, 0, stream>>>(cnt, N_NODES);
    degree_kernel<<<(N_EDGES + 255) / 256, 256, 0, stream>>>(dstp, cnt, N_EDGES);
    scan1_kernel<<<nb, 256, 0, stream>>>(cnt, rowptr, bsum, N_NODES);
    scan2_kernel<<<1, 256, 0, stream>>>(bsum, nb);
    scan3_kernel<<<nb, 256, 0, stream>>>(rowptr, bsum, cursor, N_NODES);
    csr_fill_kernel<<<(N_EDGES + 255) / 256, 256, 0, stream>>>(src, dstp, cursor,
                                                               colidx, N_EDGES);

    const int aggBlocks  = (N_NODES + 7) / 8;     // 1 wave per node
    const int gemmBlocks = (N_NODES + 127) / 128; // 1 wave per 16 rows

    // ---- Layer 1: mean-aggregate x, h = relu(mean@W1l + b1 + x@W1r) ----
    aggregate_mean_kernel<<<aggBlocks, 256, 0, stream>>>(x, rowptr, cnt, colidx,
                                                         mean, N_NODES);
    sage_gemm_kernel<<<gemmBlocks, 256, 0, stream>>>(mean, x, W1l, b1, W1r,
                                                     h, N_NODES, 1);

    // ---- Layer 2: mean-aggregate h (reuse CSR + mean buffer) ----
    aggregate_mean_kernel<<<aggBlocks, 256, 0, stream>>>(h, rowptr, cnt, colidx,
                                                         mean, N_NODES);
    sage_gemm_kernel<<<gemmBlocks, 256, 0, stream>>>(mean, h, W2l, b2, W2r,
                                                     out, N_NODES, 0);
}